// customLoss_35296041238974
// MI455X (gfx1250) — compile-verified
//
#include <hip/hip_runtime.h>
#include <stdint.h>

// Problem constants (reference: N, M, D = 1024, 1024, 256)
#define N_PTS 1024
#define M_PTS 1024
#define DIM   256

// K-chunking of the LDS panels: 4 chunks of 64 floats, padded stride 68
// (68*4B = 272B keeps 16B alignment for b128 LDS ops and skews banks).
#define KC   64
#define KCP  68
#define NCHUNK (DIM / KC)

typedef float v2f __attribute__((ext_vector_type(2)));
typedef float v8f __attribute__((ext_vector_type(8)));
typedef int   v4i_vs __attribute__((vector_size(16)));  // matches builtin's V4i

#if defined(__has_builtin)
#if __has_builtin(__builtin_amdgcn_global_load_async_to_lds_b128)
#define HAVE_ASYNC_B128 1
#endif
#if __has_builtin(__builtin_amdgcn_s_wait_asynccnt)
#define HAVE_WAIT_ASYNC 1
#endif
#endif

// Async copy of 16B global -> LDS (GLOBAL_LOAD_ASYNC_TO_LDS_B128, ASYNCcnt).
__device__ __forceinline__ void async_copy_b128(const float* g, const float* l) {
#if defined(HAVE_ASYNC_B128)
  __builtin_amdgcn_global_load_async_to_lds_b128(
      (__attribute__((address_space(1))) v4i_vs*)(uintptr_t)g,
      (__attribute__((address_space(3))) v4i_vs*)(uint32_t)(uintptr_t)l,
      /*offset=*/0, /*cpol=*/0);
#else
  // vdst = LDS byte address (low 32 bits of the generic shared pointer)
  asm volatile("global_load_async_to_lds_b128 %0, %1, off"
               :: "v"((uint32_t)(uintptr_t)l), "v"(g)
               : "memory");
#endif
}

__device__ __forceinline__ void wait_async0() {
#if defined(HAVE_WAIT_ASYNC)
  __builtin_amdgcn_s_wait_asynccnt(0);
#else
  asm volatile("s_wait_asynccnt 0" ::: "memory");
#endif
}

// Sum a per-lane scalar across all 32 lanes using v_wmma_f32_16x16x4_f32.
// Stage 1: t = A(p,0) x ones   -> row m of t = sum of the lane values mapped
//          into A's row m (constant across columns).
// Stage 2: 4x acc = ones x (t[2q],t[2q+1]) -> sums all 16 rows of t.
// Every lane-value slot is summed exactly once, so the result is invariant
// to the precise A/B VGPR striping. Result replicated in all lanes.
__device__ __forceinline__ float wave_reduce_wmma(float p) {
  v2f a;    a.x = p;    a.y = 0.0f;
  v2f ones; ones.x = 1.0f; ones.y = 1.0f;
  v8f t = {};
  t = __builtin_amdgcn_wmma_f32_16x16x4_f32(false, a, false, ones,
                                            (short)0, t, false, false);
  v8f acc = {};
#pragma unroll
  for (int q = 0; q < 4; ++q) {
    v2f b; b.x = t[2 * q]; b.y = t[2 * q + 1];
    acc = __builtin_amdgcn_wmma_f32_16x16x4_f32(false, ones, false, b,
                                                (short)0, acc, false, false);
  }
  return acc[0];
}

// Block tile: 64 rows (FA) x 64 cols (FB); 8 waves, each wave a 32x16 tile;
// each lane accumulates an 8x2 register tile over the full D=256.
__global__ __launch_bounds__(256) void l1aff_kernel(
    const float* __restrict__ FA, const float* __restrict__ FB,
    const int* __restrict__ Aff, float* __restrict__ blockOut) {
  __shared__ float FA_s[64 * KCP];
  __shared__ float FB_s[64 * KCP];
  __shared__ float wsum[8];

  const int tid  = threadIdx.x;
  const int lane = tid & 31;
  const int w    = tid >> 5;
  const int wr   = w >> 2;      // 0..1  (wave row of 32)
  const int wc   = w & 3;       // 0..3  (wave col of 16)
  const int lr   = lane >> 3;   // 0..3  (8 rows each)
  const int lc   = lane & 7;    // 0..7  (2 cols each)

  const int ibase = blockIdx.y * 64;
  const int jbase = blockIdx.x * 64;
  const int row0  = wr * 32 + lr * 8;  // local FA row of acc[0][*]
  const int col0  = wc * 16 + lc * 2;  // local FB row of acc[*][0]

  float acc[8][2];
#pragma unroll
  for (int r = 0; r < 8; ++r) { acc[r][0] = 0.f; acc[r][1] = 0.f; }

  const float4* FA4 = reinterpret_cast<const float4*>(FA_s);
  const float4* FB4 = reinterpret_cast<const float4*>(FB_s);

#pragma unroll 1
  for (int c = 0; c < NCHUNK; ++c) {
    const int k0 = c * KC;
    __syncthreads();  // previous chunk's LDS reads done before overwrite
    // Stage 64x64-float panels of FA and FB: 1024 float4 each, 4 per thread.
#pragma unroll
    for (int i = 0; i < 4; ++i) {
      const int idx = tid + i * 256;
      const int row = idx >> 4;          // 0..63
      const int kk  = (idx & 15) * 4;    // 0..60
      async_copy_b128(FA + (size_t)(ibase + row) * DIM + k0 + kk,
                      &FA_s[row * KCP + kk]);
      async_copy_b128(FB + (size_t)(jbase + row) * DIM + k0 + kk,
                      &FB_s[row * KCP + kk]);
    }
    wait_async0();
    __syncthreads();

#pragma unroll 4
    for (int kk = 0; kk < KC / 4; ++kk) {
      const float4 b0 = FB4[(col0 + 0) * (KCP / 4) + kk];
      const float4 b1 = FB4[(col0 + 1) * (KCP / 4) + kk];
#pragma unroll
      for (int r = 0; r < 8; ++r) {
        const float4 av = FA4[(row0 + r) * (KCP / 4) + kk];
        acc[r][0] += fabsf(av.x - b0.x) + fabsf(av.y - b0.y) +
                     fabsf(av.z - b0.z) + fabsf(av.w - b0.w);
        acc[r][1] += fabsf(av.x - b1.x) + fabsf(av.y - b1.y) +
                     fabsf(av.z - b1.z) + fabsf(av.w - b1.w);
      }
    }
  }

  // Loss: Aff==1 -> d, else 1/d. Per-lane partial over its 8x2 tile.
  float p = 0.f;
  const int ii0 = ibase + row0;
  const int jj0 = jbase + col0;
#pragma unroll
  for (int r = 0; r < 8; ++r) {
    const int2 af =
        *reinterpret_cast<const int2*>(Aff + (size_t)(ii0 + r) * M_PTS + jj0);
    const float d0 = acc[r][0], d1 = acc[r][1];
    p += (af.x == 1) ? d0 : 1.0f / d0;
    p += (af.y == 1) ? d1 : 1.0f / d1;
  }

  // Cross-lane sum via WMMA (EXEC all ones here: no divergence above).
  const float wave_total = wave_reduce_wmma(p);

  if (lane == 0) wsum[w] = wave_total;
  __syncthreads();
  if (tid == 0) {
    float s = 0.f;
#pragma unroll
    for (int q = 0; q < 8; ++q) s += wsum[q];
    blockOut[blockIdx.y * gridDim.x + blockIdx.x] = s;
  }
}

// Deterministic final reduction of the 256 block partials.
__global__ __launch_bounds__(256) void reduce256_kernel(
    const float* __restrict__ in, float* __restrict__ out) {
  __shared__ float s[8];
  const int t = threadIdx.x;
  float v = in[t];
#pragma unroll
  for (int off = 16; off > 0; off >>= 1) v += __shfl_down(v, off, 32);
  if ((t & 31) == 0) s[t >> 5] = v;
  __syncthreads();
  if (t == 0) {
    float r = 0.f;
#pragma unroll
    for (int q = 0; q < 8; ++q) r += s[q];
    out[0] = r;
  }
}

extern "C" void kernel_launch(void* const* d_in, const int* in_sizes, int n_in,
                              void* d_out, int out_size, void* d_ws,
                              size_t ws_size, hipStream_t stream) {
  (void)in_sizes; (void)n_in; (void)out_size; (void)ws_size;
  const float* FA  = (const float*)d_in[0];
  const float* FB  = (const float*)d_in[1];
  const int*   Aff = (const int*)d_in[2];
  float* blockOut  = (float*)d_ws;  // 256 floats of scratch

  dim3 grid(M_PTS / 64, N_PTS / 64);  // 16 x 16 = 256 blocks
  l1aff_kernel<<<grid, 256, 0, stream>>>(FA, FB, Aff, blockOut);
  reduce256_kernel<<<1, 256, 0, stream>>>(blockOut, (float*)d_out);
}